// SuperPointNet_48095043780758
// MI455X (gfx1250) — compile-verified
//
#include <hip/hip_runtime.h>
#include <hip/hip_bf16.h>
#include <math.h>

typedef __attribute__((ext_vector_type(2))) float v2f;
typedef __attribute__((ext_vector_type(8))) float v8f;

#define DEVINL __device__ __forceinline__

static DEVINL int imin(int a, int b) { return a < b ? a : b; }

// ---------------------------------------------------------------------------
// Generic WMMA fp32 GEMM:  Y[M,NC] = A[M,K] @ W[K,NC] (+bias)
// One wave computes one 16x16 output tile via V_WMMA_F32_16X16X4_F32.
// Inner loop is branch-free: unconditional main loop + clamped/select tail.
// MODE 0: dense A.
// MODE 1: encoder gather: A row r -> (q = r/ns, neighbor j), columns are
//         [rel_xyz(3), x_prev[nidx[r]]] with rel = p0[nidx*sp] - p0[q*sc].
//         Requires K >= 4 (true: K in {35,67,131,259}).
// MODE 2: A row = concat(A1[r,0:K1], rowvec[0:K-K1]); K1%4==0, K%4==0.
// ---------------------------------------------------------------------------
template <int MODE>
__global__ __launch_bounds__(256) void gemm_wmma_f32(
    const float* __restrict__ A, const float* __restrict__ W,
    const float* __restrict__ bias, float* __restrict__ Y,
    int M, int K, int NC,
    const float* __restrict__ p0, const float* __restrict__ xprev,
    const int* __restrict__ nidx, int sp, int sc, int ns,
    int K1, const float* __restrict__ rowvec)
{
    const int lane = threadIdx.x & 31;
    const int wave = threadIdx.x >> 5;
    const int tilesN = (NC + 15) >> 4;
    const int tilesM = (M + 15) >> 4;
    const int tile = blockIdx.x * 8 + wave;
    if (tile >= tilesM * tilesN) return;   // wave-uniform: EXEC stays all-ones
    const int tm = tile / tilesN;
    const int tn = tile - tm * tilesN;

    const int lrow = lane & 15;
    const int half = lane >> 4;   // half 0 -> K = k0,k0+1 ; half 1 -> k0+2,k0+3
    const int kh = half * 2;

    int row = tm * 16 + lrow;
    if (row >= M) row = M - 1;            // clamp; stores are guarded
    const int col = tn * 16 + lrow;
    const int colc = (col < NC) ? col : (NC - 1);

    // B-panel pointer: this lane reads W[k0+kh][colc] and W[k0+kh+1][colc].
    const float* pW = W + (size_t)kh * NC + colc;
    const size_t stepW = (size_t)4 * NC;
    __builtin_prefetch(pW, 0, 1);  // global_prefetch of weight panel

    v8f acc = {0.f, 0.f, 0.f, 0.f, 0.f, 0.f, 0.f, 0.f};
    int k0 = 0;

    const float* arow = nullptr;
    const float* xb = nullptr;
    const float* pA = nullptr;

    if (MODE == 1) {
        const int q = row / ns;
        const int ridx = nidx[row];
        const float* pr = p0 + (size_t)ridx * sp * 3;
        const float* pq = p0 + (size_t)q * sc * 3;
        const float rel0 = pr[0] - pq[0];
        const float rel1 = pr[1] - pq[1];
        const float rel2 = pr[2] - pq[2];
        xb = xprev + (size_t)ridx * (size_t)(K - 3);
        // First K-step (k = 0..3): [rel0, rel1 | rel2, xb[0]] -- unconditional.
        {
            v2f av = {half ? rel2 : rel0, half ? xb[0] : rel1};
            v2f bv = {pW[0], pW[NC]};
            acc = __builtin_amdgcn_wmma_f32_16x16x4_f32(
                false, av, false, bv, (short)0, acc, false, false);
            pW += stepW;
        }
        pA = xb + 1 + kh;  // feature index for k = 4+kh
        k0 = 4;
    } else {
        arow = A + (size_t)row * (size_t)((MODE == 2) ? K1 : K);
        pA = arow + kh;
    }

    if (MODE == 2) {
        // Dense part [0, K1) then broadcast row [K1, K); both multiples of 4.
        for (; k0 < K1; k0 += 4) {
            v2f av = {pA[0], pA[1]};
            v2f bv = {pW[0], pW[NC]};
            acc = __builtin_amdgcn_wmma_f32_16x16x4_f32(
                false, av, false, bv, (short)0, acc, false, false);
            pA += 4; pW += stepW;
        }
        const float* pR = rowvec + kh;
        for (; k0 < K; k0 += 4) {
            v2f av = {pR[0], pR[1]};
            v2f bv = {pW[0], pW[NC]};
            acc = __builtin_amdgcn_wmma_f32_16x16x4_f32(
                false, av, false, bv, (short)0, acc, false, false);
            pR += 4; pW += stepW;
        }
    } else {
        const int kmain = K & ~3;
        for (; k0 < kmain; k0 += 4) {
            v2f av = {pA[0], pA[1]};
            v2f bv = {pW[0], pW[NC]};
            acc = __builtin_amdgcn_wmma_f32_16x16x4_f32(
                false, av, false, bv, (short)0, acc, false, false);
            pA += 4; pW += stepW;
        }
        if (kmain < K) {
            // Tail: unconditional loads at clamped (valid) addresses,
            // zeroed with selects -> no EXEC manipulation.
            const int ka = kmain + kh;
            const int kb = ka + 1;
            const int kac = imin(ka, K - 1);
            const int kbc = imin(kb, K - 1);
            float va, vb;
            if (MODE == 1) {
                va = xb[kac - 3];
                vb = xb[kbc - 3];
            } else {
                va = arow[kac];
                vb = arow[kbc];
            }
            const float wa = W[(size_t)kac * NC + colc];
            const float wb = W[(size_t)kbc * NC + colc];
            v2f av = {(ka < K) ? va : 0.f, (kb < K) ? vb : 0.f};
            v2f bv = {(ka < K) ? wa : 0.f, (kb < K) ? wb : 0.f};
            acc = __builtin_amdgcn_wmma_f32_16x16x4_f32(
                false, av, false, bv, (short)0, acc, false, false);
        }
    }

    if (col < NC) {
        const float bb = bias ? bias[col] : 0.f;
        const int rbase = tm * 16 + half * 8;
        float* yp = Y + (size_t)rbase * NC + col;
        if (rbase + 8 <= M) {
#pragma unroll
            for (int r = 0; r < 8; ++r) yp[(size_t)r * NC] = acc[r] + bb;
        } else {
#pragma unroll
            for (int r = 0; r < 8; ++r)
                if (rbase + r < M) yp[(size_t)r * NC] = acc[r] + bb;
        }
    }
}

// ---------------------------------------------------------------------------
// Streaming kNN: one thread per query; reference tiles staged into LDS with
// CDNA5 async VMEM->LDS loads (GLOBAL_LOAD_ASYNC_TO_LDS_B32, ASYNCcnt).
// Register-resident ascending top-K matches lax.top_k on -dist (stable ties).
// Points are strided views of p0 (coarse sets are p0[i*stride]).
// ---------------------------------------------------------------------------
template <int K>
__global__ __launch_bounds__(256) void knn_kernel(
    const float* __restrict__ p0, int qstride, int rstride,
    int Mq, int Mr, int* __restrict__ outIdx)
{
    __shared__ float sref[256 * 3];
    const int q = blockIdx.x * blockDim.x + threadIdx.x;
    const bool act = q < Mq;
    const int qq = act ? q : 0;
    const float qx = p0[(size_t)qq * qstride * 3 + 0];
    const float qy = p0[(size_t)qq * qstride * 3 + 1];
    const float qz = p0[(size_t)qq * qstride * 3 + 2];

    float bd[K];
    int bi[K];
#pragma unroll
    for (int t = 0; t < K; ++t) { bd[t] = 3.0e38f; bi[t] = 0; }

    for (int base = 0; base < Mr; base += 256) {
        const int n = (Mr - base < 256) ? (Mr - base) : 256;
        __syncthreads();  // previous tile fully consumed
        // Async-copy n*3 floats of (strided) reference points into LDS.
        for (int e = threadIdx.x; e < n * 3; e += 256) {
            const int j = e / 3;
            const int c = e - 3 * j;
            const float* ga = p0 + ((size_t)(base + j) * rstride * 3 + c);
            const unsigned lo = (unsigned)(uintptr_t)(&sref[e]);
            asm volatile("global_load_async_to_lds_b32 %0, %1, off"
                         :: "v"(lo), "v"(ga) : "memory");
        }
        asm volatile("s_wait_asynccnt 0x0" ::: "memory");
        __syncthreads();
        for (int j = 0; j < n; ++j) {
            const float dx = qx - sref[j * 3 + 0];
            const float dy = qy - sref[j * 3 + 1];
            const float dz = qz - sref[j * 3 + 2];
            const float d = dx * dx + dy * dy + dz * dz;
            if (d < bd[K - 1]) {
                bd[K - 1] = d; bi[K - 1] = base + j;
#pragma unroll
                for (int s = K - 1; s > 0; --s) {
                    if (bd[s] < bd[s - 1]) {
                        float td = bd[s]; bd[s] = bd[s - 1]; bd[s - 1] = td;
                        int ti = bi[s]; bi[s] = bi[s - 1]; bi[s - 1] = ti;
                    }
                }
            }
        }
    }
    if (act) {
#pragma unroll
        for (int t = 0; t < K; ++t) outIdx[(size_t)q * K + t] = bi[t];
    }
}

// ---------------------------------------------------------------------------
// Train-mode BN statistics: per-channel sum & sumsq via LDS atomics, one
// global atomic per (block, channel). C must be a power of two (it is for
// every BN layer in this net).
// ---------------------------------------------------------------------------
__global__ __launch_bounds__(256) void bn_stats(
    const float* __restrict__ Y, int R, int C, int rowsPerBlock,
    float* __restrict__ gsum, float* __restrict__ gsq)
{
    __shared__ float ls[512];
    __shared__ float lq[512];
    for (int c = threadIdx.x; c < C; c += 256) { ls[c] = 0.f; lq[c] = 0.f; }
    __syncthreads();
    const unsigned cmask = (unsigned)C - 1u;
    size_t r0 = (size_t)blockIdx.x * rowsPerBlock;
    size_t r1 = r0 + rowsPerBlock;
    if (r1 > (size_t)R) r1 = (size_t)R;
    const size_t e1 = r1 * C;
    for (size_t e = r0 * C + threadIdx.x; e < e1; e += 256) {
        const float v = Y[e];
        const int c = (int)((unsigned)e & cmask);
        atomicAdd(&ls[c], v);
        atomicAdd(&lq[c], v * v);
    }
    __syncthreads();
    for (int c = threadIdx.x; c < C; c += 256) {
        atomicAdd(&gsum[c], ls[c]);
        atomicAdd(&gsq[c], lq[c]);
    }
}

__global__ __launch_bounds__(256) void bn_apply_relu(
    const float* __restrict__ Y, const float* __restrict__ gsum,
    const float* __restrict__ gsq, const float* __restrict__ g,
    const float* __restrict__ beta, int R, int C, float* __restrict__ X)
{
    const size_t e = (size_t)blockIdx.x * blockDim.x + threadIdx.x;
    if (e >= (size_t)R * C) return;
    const int c = (int)((unsigned)e & ((unsigned)C - 1u));
    const float inv = 1.0f / (float)R;
    const float m = gsum[c] * inv;
    const float var = gsq[c] * inv - m * m;
    const float v = (Y[e] - m) * rsqrtf(var + 1e-5f) * g[c] + beta[c];
    X[e] = v > 0.f ? v : 0.f;
}

// BN + ReLU + max-pool over the ns neighbor axis (encoder stages).
__global__ __launch_bounds__(256) void bn_relu_maxpool(
    const float* __restrict__ Y, const float* __restrict__ gsum,
    const float* __restrict__ gsq, const float* __restrict__ g,
    const float* __restrict__ beta, int Mq, int ns, int C,
    float* __restrict__ X)
{
    const int e = blockIdx.x * blockDim.x + threadIdx.x;
    if (e >= Mq * C) return;
    const int q = e / C;
    const int c = e - q * C;
    const float inv = 1.0f / (float)(Mq * ns);
    const float m = gsum[c] * inv;
    const float var = gsq[c] * inv - m * m;
    const float s = rsqrtf(var + 1e-5f) * g[c];
    const float bb = beta[c];
    const float* yb = Y + (size_t)q * ns * C + c;
    float best = 0.f;  // relu outputs are >= 0
    for (int j = 0; j < ns; ++j) {
        const float v = (yb[(size_t)j * C] - m) * s + bb;
        best = fmaxf(best, v);
    }
    X[e] = best;
}

// Column mean over rows (dec5 global pooling).
__global__ __launch_bounds__(256) void col_mean(
    const float* __restrict__ X, int R, int C, float* __restrict__ out)
{
    const int c = blockIdx.x * blockDim.x + threadIdx.x;
    if (c >= C) return;
    float s = 0.f;
    for (int r = 0; r < R; ++r) s += X[(size_t)r * C + c];
    out[c] = s / (float)R;
}

// Small GEMV + bias + ReLU (dec5 l2 on the 1xC mean vector).
__global__ __launch_bounds__(256) void gemv_bias_relu(
    const float* __restrict__ x, const float* __restrict__ W,
    const float* __restrict__ b, int K, int C, float* __restrict__ out)
{
    const int o = blockIdx.x * blockDim.x + threadIdx.x;
    if (o >= C) return;
    float s = b[o];
    for (int k = 0; k < K; ++k) s += x[k] * W[(size_t)k * C + o];
    out[o] = fmaxf(s, 0.f);
}

// out[m,c] = a[m,c] + sum_k w_k * b[idx[m,k],c]   (3-NN inv-sq-dist interp)
__global__ __launch_bounds__(256) void combine_interp(
    const float* __restrict__ a, const float* __restrict__ b,
    const int* __restrict__ idx, const float* __restrict__ p0,
    int fstride, int cstride, int Mf, int C, float* __restrict__ out)
{
    const size_t e = (size_t)blockIdx.x * blockDim.x + threadIdx.x;
    if (e >= (size_t)Mf * C) return;
    const int m = (int)(e / (size_t)C);
    const int c = (int)(e - (size_t)m * C);
    const float px = p0[(size_t)m * fstride * 3 + 0];
    const float py = p0[(size_t)m * fstride * 3 + 1];
    const float pz = p0[(size_t)m * fstride * 3 + 2];
    float w[3]; int id[3]; float sw = 0.f;
#pragma unroll
    for (int k = 0; k < 3; ++k) {
        id[k] = idx[(size_t)m * 3 + k];
        const float dx = px - p0[(size_t)id[k] * cstride * 3 + 0];
        const float dy = py - p0[(size_t)id[k] * cstride * 3 + 1];
        const float dz = pz - p0[(size_t)id[k] * cstride * 3 + 2];
        w[k] = 1.0f / (dx * dx + dy * dy + dz * dz + 1e-8f);
        sw += w[k];
    }
    float r = a[e];
#pragma unroll
    for (int k = 0; k < 3; ++k)
        r += (w[k] / sw) * b[(size_t)id[k] * C + c];
    out[e] = r;
}

__global__ __launch_bounds__(256) void build_xin(
    const float* __restrict__ p0, const float* __restrict__ x0,
    float* __restrict__ x, int N)
{
    const int i = blockIdx.x * blockDim.x + threadIdx.x;
    if (i >= N) return;
#pragma unroll
    for (int d = 0; d < 3; ++d) {
        x[(size_t)i * 6 + d] = p0[(size_t)i * 3 + d];
        x[(size_t)i * 6 + 3 + d] = x0[(size_t)i * 3 + d];
    }
}

// ---------------------------------------------------------------------------
// Host orchestration
// ---------------------------------------------------------------------------
extern "C" void kernel_launch(void* const* d_in, const int* in_sizes, int n_in,
                              void* d_out, int out_size, void* d_ws,
                              size_t ws_size, hipStream_t stream)
{
    (void)in_sizes; (void)n_in; (void)out_size; (void)ws_size;
    const float* P0 = (const float*)d_in[0];
    const float* X0 = (const float*)d_in[1];
    // d_in[2] = o0 (unused, batch==1)
    auto f = [&](int i) { return (const float*)d_in[i]; };
    // params in setup_inputs() insertion order:
    const float *enc1W = f(3), *enc1g = f(4), *enc1b = f(5);
    const float *enc2W = f(6), *enc2g = f(7), *enc2b = f(8);
    const float *enc3W = f(9), *enc3g = f(10), *enc3b = f(11);
    const float *enc4W = f(12), *enc4g = f(13), *enc4b = f(14);
    const float *enc5W = f(15), *enc5g = f(16), *enc5b = f(17);
    const float *d5l1W = f(18), *d5l1b = f(19), *d5l1g = f(20), *d5l1be = f(21);
    const float *d5l2W = f(22), *d5l2b = f(23);
    const float *d4l1W = f(24), *d4l1b = f(25), *d4l1g = f(26), *d4l1be = f(27);
    const float *d4l2W = f(28), *d4l2b = f(29), *d4l2g = f(30), *d4l2be = f(31);
    const float *d3l1W = f(32), *d3l1b = f(33), *d3l1g = f(34), *d3l1be = f(35);
    const float *d3l2W = f(36), *d3l2b = f(37), *d3l2g = f(38), *d3l2be = f(39);
    const float *d2l1W = f(40), *d2l1b = f(41), *d2l1g = f(42), *d2l1be = f(43);
    const float *d2l2W = f(44), *d2l2b = f(45), *d2l2g = f(46), *d2l2be = f(47);
    const float *d1l1W = f(48), *d1l1b = f(49), *d1l1g = f(50), *d1l1be = f(51);
    const float *d1l2W = f(52), *d1l2b = f(53), *d1l2g = f(54), *d1l2be = f(55);
    const float *cl1W = f(56), *cl1b = f(57), *cl1g = f(58), *cl1be = f(59);
    const float *cl2W = f(60), *cl2b = f(61);

    // ---- workspace layout -------------------------------------------------
    char* ws = (char*)d_ws;
    size_t off = 0;
    auto alloc = [&](size_t nf) -> float* {
        off = (off + 255) & ~(size_t)255;
        float* p = (float*)(ws + off);
        off += nf * sizeof(float);
        return p;
    };
    float* XIN = alloc((size_t)65536 * 6);
    float* X1 = alloc((size_t)65536 * 32);
    float* X2 = alloc((size_t)16384 * 64);
    float* X3 = alloc((size_t)4096 * 128);
    float* X4 = alloc((size_t)1024 * 256);
    float* X5 = alloc((size_t)256 * 512);
    float* Yb = alloc((size_t)16384 * 16 * 64);  // largest pre-activation (67MB)
    float* ABUF = alloc((size_t)65536 * 32);
    float* BBUF = alloc((size_t)16384 * 32);
    int* IDX = (int*)alloc((size_t)16384 * 16);
    float* SSUM = alloc(512);
    float* SSQ = alloc(512);
    float* MEANV = alloc(512);
    float* GFEAT = alloc(512);

    auto cdiv = [](int a, int b) { return (a + b - 1) / b; };
    auto stats = [&](const float* Y, int R, int C) {
        hipMemsetAsync(SSUM, 0, 2 * 512 * sizeof(float), stream);
        hipLaunchKernelGGL(bn_stats, dim3(cdiv(R, 64)), dim3(256), 0, stream,
                           Y, R, C, 64, SSUM, SSQ);
    };
    auto gemm0 = [&](const float* A, const float* W, const float* bias,
                     float* Y, int M, int K, int NC) {
        const int tiles = cdiv(M, 16) * cdiv(NC, 16);
        hipLaunchKernelGGL((gemm_wmma_f32<0>), dim3(cdiv(tiles, 8)), dim3(256),
                           0, stream, A, W, bias, Y, M, K, NC,
                           nullptr, nullptr, nullptr, 0, 0, 1, 0, nullptr);
    };
    auto gemm_gather = [&](const float* W, float* Y, int Mq, int ns, int K,
                           int NC, const float* xprev, int sp, int sc) {
        const int M = Mq * ns;
        const int tiles = cdiv(M, 16) * cdiv(NC, 16);
        hipLaunchKernelGGL((gemm_wmma_f32<1>), dim3(cdiv(tiles, 8)), dim3(256),
                           0, stream, nullptr, W, nullptr, Y, M, K, NC,
                           P0, xprev, IDX, sp, sc, ns, 0, nullptr);
    };
    auto apply = [&](const float* Y, const float* g, const float* be, int R,
                     int C, float* X) {
        hipLaunchKernelGGL(bn_apply_relu, dim3(cdiv(R * C, 256)), dim3(256), 0,
                           stream, Y, SSUM, SSQ, g, be, R, C, X);
    };

    // ---- encoder ----------------------------------------------------------
    hipLaunchKernelGGL(build_xin, dim3(cdiv(65536, 256)), dim3(256), 0, stream,
                       P0, X0, XIN, 65536);
    // enc1: 6 -> 32, stride 1 (no gather, no pool)
    gemm0(XIN, enc1W, nullptr, Yb, 65536, 6, 32);
    stats(Yb, 65536, 32);
    apply(Yb, enc1g, enc1b, 65536, 32, X1);

    struct Enc { int Mq, Mr, sp, sc, K, NC; const float *W, *g, *b, *xp; float* Xo; };
    const Enc encs[4] = {
        {16384, 65536, 1, 4, 35, 64, enc2W, enc2g, enc2b, X1, X2},
        {4096, 16384, 4, 16, 67, 128, enc3W, enc3g, enc3b, X2, X3},
        {1024, 4096, 16, 64, 131, 256, enc4W, enc4g, enc4b, X3, X4},
        {256, 1024, 64, 256, 259, 512, enc5W, enc5g, enc5b, X4, X5}};
    for (int i = 0; i < 4; ++i) {
        const Enc& E = encs[i];
        hipLaunchKernelGGL((knn_kernel<16>), dim3(cdiv(E.Mq, 256)), dim3(256),
                           0, stream, P0, E.sc, E.sp, E.Mq, E.Mr, IDX);
        gemm_gather(E.W, Yb, E.Mq, 16, E.K, E.NC, E.xp, E.sp, E.sc);
        stats(Yb, E.Mq * 16, E.NC);
        hipLaunchKernelGGL(bn_relu_maxpool, dim3(cdiv(E.Mq * E.NC, 256)),
                           dim3(256), 0, stream, Yb, SSUM, SSQ, E.g, E.b,
                           E.Mq, 16, E.NC, E.Xo);
    }

    // ---- dec5 head --------------------------------------------------------
    hipLaunchKernelGGL(col_mean, dim3(2), dim3(256), 0, stream, X5, 256, 512, MEANV);
    hipLaunchKernelGGL(gemv_bias_relu, dim3(2), dim3(256), 0, stream, MEANV,
                       d5l2W, d5l2b, 512, 512, GFEAT);
    {   // concat(x5, gfeat) [256,1024] @ W -> BN -> ReLU -> X5 (overwrite)
        const int tiles = cdiv(256, 16) * cdiv(512, 16);
        hipLaunchKernelGGL((gemm_wmma_f32<2>), dim3(cdiv(tiles, 8)), dim3(256),
                           0, stream, X5, d5l1W, d5l1b, Yb, 256, 1024, 512,
                           nullptr, nullptr, nullptr, 0, 0, 1, 512, GFEAT);
        stats(Yb, 256, 512);
        apply(Yb, d5l1g, d5l1be, 256, 512, X5);
    }

    // ---- decoder up-stages ------------------------------------------------
    struct Dec {
        int Mf, Mc, sf, scc, Cf, Cc;  // Cf = channel out, Cc = coarse in
        const float *l1W, *l1b, *l1g, *l1be, *l2W, *l2b, *l2g, *l2be;
        float *Xf, *Xc;
    };
    const Dec decs[4] = {
        {1024, 256, 64, 256, 256, 512, d4l1W, d4l1b, d4l1g, d4l1be,
         d4l2W, d4l2b, d4l2g, d4l2be, X4, X5},
        {4096, 1024, 16, 64, 128, 256, d3l1W, d3l1b, d3l1g, d3l1be,
         d3l2W, d3l2b, d3l2g, d3l2be, X3, X4},
        {16384, 4096, 4, 16, 64, 128, d2l1W, d2l1b, d2l1g, d2l1be,
         d2l2W, d2l2b, d2l2g, d2l2be, X2, X3},
        {65536, 16384, 1, 4, 32, 64, d1l1W, d1l1b, d1l1g, d1l1be,
         d1l2W, d1l2b, d1l2g, d1l2be, X1, X2}};
    for (int i = 0; i < 4; ++i) {
        const Dec& D = decs[i];
        // a = relu(bn(x_f @ l1W + b))
        gemm0(D.Xf, D.l1W, D.l1b, Yb, D.Mf, D.Cf, D.Cf);
        stats(Yb, D.Mf, D.Cf);
        apply(Yb, D.l1g, D.l1be, D.Mf, D.Cf, ABUF);
        // b = relu(bn(x_c @ l2W + b))
        gemm0(D.Xc, D.l2W, D.l2b, Yb, D.Mc, D.Cc, D.Cf);
        stats(Yb, D.Mc, D.Cf);
        apply(Yb, D.l2g, D.l2be, D.Mc, D.Cf, BBUF);
        // 3-NN interpolation coarse -> fine, add to a, overwrite skip buffer
        hipLaunchKernelGGL((knn_kernel<3>), dim3(cdiv(D.Mf, 256)), dim3(256),
                           0, stream, P0, D.sf, D.scc, D.Mf, D.Mc, IDX);
        hipLaunchKernelGGL(combine_interp, dim3(cdiv(D.Mf * D.Cf, 256)),
                           dim3(256), 0, stream, ABUF, BBUF, IDX, P0, D.sf,
                           D.scc, D.Mf, D.Cf, D.Xf);
    }

    // ---- classifier -------------------------------------------------------
    gemm0(X1, cl1W, cl1b, Yb, 65536, 32, 32);
    stats(Yb, 65536, 32);
    apply(Yb, cl1g, cl1be, 65536, 32, ABUF);
    gemm0(ABUF, cl2W, cl2b, (float*)d_out, 65536, 32, 13);
}